// YoloLayer_34497177321903
// MI455X (gfx1250) — compile-verified
//
#include <hip/hip_runtime.h>
#include <math.h>

#define NB     16
#define NAV    3
#define NC     80
#define NH     76
#define NW     76
#define MAXGT  50
#define PLANE  (NH*NW)            // 5776
#define CELLS_PER_B (NAV*PLANE)   // 17328
#define MAIN_BLOCKS 68            // ceil(17328/256)
#define SCL    0.125f             // nW / IMG = 76/608
#define SILTH  0.6f

typedef __attribute__((ext_vector_type(2))) float v2f;
typedef __attribute__((ext_vector_type(8))) float v8f;

__constant__ float c_aw[3] = {1.5f, 2.375f, 5.0f};   // ANCHORS/8 widths
__constant__ float c_ah[3] = {2.0f, 4.5f, 3.5f};     // ANCHORS/8 heights

__device__ __forceinline__ float iou_cwh(float x1, float y1, float w1, float h1,
                                         float x2, float y2, float w2, float h2) {
    float mx = fminf(x1 - w1*0.5f, x2 - w2*0.5f);
    float Mx = fmaxf(x1 + w1*0.5f, x2 + w2*0.5f);
    float my = fminf(y1 - h1*0.5f, y2 - h2*0.5f);
    float My = fmaxf(y1 + h1*0.5f, y2 + h2*0.5f);
    float cw = w1 + w2 - (Mx - mx);
    float ch = h1 + h2 - (My - my);
    float carea = (cw <= 0.0f || ch <= 0.0f) ? 0.0f : cw * ch;
    float uarea = w1*h1 + w2*h2 - carea;
    return carea / uarea;
}

// Wave32 sum using the matrix pipe: V_WMMA_F32_16X16X4_F32 with B = ones.
// A layout (ISA 7.12.2, 32-bit A 16x4): lanes 0-15 -> A[L][0..1], lanes 16-31 -> A[L-16][2..3].
// With a=(v,0): rowsum_m = v_m + v_{m+16}; D broadcasts rowsums across columns.
// Lane n<16 holds D[0..7][n]; lane n+16 holds D[8..15][n]; so sum8 + shfl_xor(16) = full sum.
__device__ __forceinline__ float wave_sum_wmma(float v) {
    v2f a; a.x = v;    a.y = 0.0f;
    v2f b; b.x = 1.0f; b.y = 1.0f;
    v8f c = {};
    c = __builtin_amdgcn_wmma_f32_16x16x4_f32(false, a, false, b, (short)0, c, false, false);
    float s = c[0] + c[1] + c[2] + c[3] + c[4] + c[5] + c[6] + c[7];
    s += __shfl_xor(s, 16, 32);
    return s;
}

// ---- Kernel 1: per-GT prep (16 blocks x 64 thr; t = threadIdx) ------------
// gtf[b][t][8] = {gx,gy,gw,gh, tx,ty,tw,th};  gtk[b][t] = key or -1
// key = gi | gj<<8 | bestn<<16 | gcls<<19
__global__ void prep_kernel(const float* __restrict__ target,
                            float* __restrict__ gtf, int* __restrict__ gtk) {
    int b = blockIdx.x;
    int t = threadIdx.x;
    if (t >= MAXGT) return;
    const float* tb = target + (size_t)b * MAXGT * 5;
    const float* tg = tb + t * 5;
    float cls = tg[0];
    float gx = tg[1] * SCL, gy = tg[2] * SCL, gw = tg[3] * SCL, gh = tg[4] * SCL;
    bool valid = true;                       // cumprod(x != 0) up to t
    for (int u = 0; u <= t; ++u) valid = valid && (tb[u*5 + 1] != 0.0f);
    int bestn = 0; float best = -1.0f;       // first-max argmax (strict >)
    for (int n = 0; n < 3; ++n) {
        float v = iou_cwh(0.f, 0.f, gw, gh, 0.f, 0.f, c_aw[n], c_ah[n]);
        if (v > best) { best = v; bestn = n; }
    }
    int gi = (int)gx, gj = (int)gy;          // positive -> trunc == floor
    float* o = gtf + ((size_t)b * MAXGT + t) * 8;
    o[0] = gx; o[1] = gy; o[2] = gw; o[3] = gh;
    o[4] = gx - (float)gi;                   // tx
    o[5] = gy - (float)gj;                   // ty
    o[6] = logf(gw / c_aw[bestn]);           // tw
    o[7] = logf(gh / c_ah[bestn]);           // th
    gtk[b * MAXGT + t] = valid ? (gi | (gj << 8) | (bestn << 16) | (((int)cls) << 19)) : -1;
}

// ---- Kernel 2: fused per-cell loss (grid: 68 x 16, 256 thr) ---------------
__global__ __launch_bounds__(256) void yolo_main_kernel(
        const float* __restrict__ out,
        const float* __restrict__ gtf,
        const int*   __restrict__ gtk,
        float* __restrict__ partials) {
    __shared__ float s_box[MAXGT][4];
    __shared__ int   s_key[MAXGT];
    __shared__ float s_wave[8];

    const int b   = blockIdx.y;
    const int tid = threadIdx.x;
    if (tid < MAXGT) {
        const float* g = gtf + ((size_t)b * MAXGT + tid) * 8;
        s_box[tid][0] = g[0]; s_box[tid][1] = g[1];
        s_box[tid][2] = g[2]; s_box[tid][3] = g[3];
        s_key[tid] = gtk[b * MAXGT + tid];
    }
    __syncthreads();

    const int cell = blockIdx.x * blockDim.x + tid;
    float loss = 0.0f;
    if (cell < CELLS_PER_B) {
        const int a = cell / PLANE;
        const int r = cell - a * PLANE;
        const int j = r / NW;
        const int i = r - j * NW;
        const float* base = out + ((size_t)(b * NAV + a)) * (5 + NC) * PLANE + r;
        float o0 = base[0 * PLANE];
        float o1 = base[1 * PLANE];
        float o2 = base[2 * PLANE];
        float o3 = base[3 * PLANE];
        float o4 = base[4 * PLANE];
        float xs   = 1.0f / (1.0f + expf(-o0));
        float ys   = 1.0f / (1.0f + expf(-o1));
        float conf = 1.0f / (1.0f + expf(-o4));
        float pxc = xs + (float)i;
        float pyc = ys + (float)j;
        float pwc = expf(o2) * c_aw[a];
        float phc = expf(o3) * c_ah[a];

        // Scan GT boxes: max IoU (silence mask) + scatter lookup (last match wins)
        float maxiou = 0.0f;
        int slot = -1;
        const int mykey = i | (j << 8) | (a << 16);
        for (int t = 0; t < MAXGT; ++t) {
            int k = s_key[t];
            if (k < 0) continue;
            float v = iou_cwh(pxc, pyc, pwc, phc,
                              s_box[t][0], s_box[t][1], s_box[t][2], s_box[t][3]);
            maxiou = fmaxf(maxiou, v);
            if ((k & 0x7FFFF) == mykey) slot = t;
        }

        if (slot >= 0) {
            const float* g = gtf + ((size_t)b * MAXGT + slot) * 8;
            float tconf = iou_cwh(g[0], g[1], g[2], g[3], pxc, pyc, pwc, phc);
            float dx = xs - g[4], dy = ys - g[5], dw = o2 - g[6], dh = o3 - g[7];
            float dc = conf - tconf;
            loss = 0.5f * (dx*dx + dy*dy + dw*dw + dh*dh) + 2.5f * dc * dc; // 0.5*OBJ_SCALE
            // class cross-entropy via online logsumexp over 80 logits
            const int gcls = s_key[slot] >> 19;
            const float* cb = base + 5 * PLANE;
            float m = -INFINITY, s = 0.0f, sel = 0.0f;
            for (int c = 0; c < NC; ++c) {
                float l  = cb[(size_t)c * PLANE];
                float nm = fmaxf(m, l);
                s = s * expf(m - nm) + expf(l - nm);
                m = nm;
                if (c == gcls) sel = l;
            }
            loss += (m + logf(s)) - sel;
        } else {
            float cm = (maxiou > SILTH) ? 0.0f : 1.0f;  // NOOBJ_SCALE = 1
            float dx = xs - 0.5f, dy = ys - 0.5f;
            loss = 0.5f * (dx*dx + dy*dy + o2*o2 + o3*o3) + 0.5f * cm * conf * conf;
        }
    }

    // Per-wave reduction on the matrix pipe (all 256 lanes active -> EXEC all ones)
    float wsum = wave_sum_wmma(loss);
    const int lane = tid & 31, wave = tid >> 5;
    if (lane == 0) s_wave[wave] = wsum;
    __syncthreads();
    if (tid == 0) {
        float tot = 0.0f;
        for (int w = 0; w < 8; ++w) tot += s_wave[w];
        partials[blockIdx.y * gridDim.x + blockIdx.x] = tot;
    }
}

// ---- Kernel 3: deterministic final reduction (1 block x 32 thr) -----------
__global__ void final_kernel(const float* __restrict__ partials, int n,
                             float* __restrict__ outp) {
    float s = 0.0f;
    for (int idx = threadIdx.x; idx < n; idx += 32) s += partials[idx];
    float tot = wave_sum_wmma(s);
    if (threadIdx.x == 0) outp[0] = tot;
}

extern "C" void kernel_launch(void* const* d_in, const int* in_sizes, int n_in,
                              void* d_out, int out_size, void* d_ws, size_t ws_size,
                              hipStream_t stream) {
    const float* output = (const float*)d_in[0];
    const float* target = (const float*)d_in[1];

    float* gtf      = (float*)d_ws;                       // 16*50*8 floats
    int*   gtk      = (int*)(gtf + NB * MAXGT * 8);       // 16*50 ints
    float* partials = (float*)(gtk + NB * MAXGT);         // 16*68 floats

    prep_kernel<<<dim3(NB), dim3(64), 0, stream>>>(target, gtf, gtk);
    yolo_main_kernel<<<dim3(MAIN_BLOCKS, NB), dim3(256), 0, stream>>>(output, gtf, gtk, partials);
    final_kernel<<<dim3(1), dim3(32), 0, stream>>>(partials, NB * MAIN_BLOCKS, (float*)d_out);
}